// RNNDecoder_82703890252496
// MI455X (gfx1250) — compile-verified
//
#include <hip/hip_runtime.h>
#include <hip/hip_bf16.h>
#include <cmath>

// ---------------------------------------------------------------------------
// GRU decoder restructured for MI455X (gfx1250):
//   phase 1: setup (h0, gi)                          - tiny
//   phase 2: 64 serial GRU steps, store every h_t    - latency bound, small
//   phase 3: ONE GEMM  H(64x1024) @ W_vocab(1024xV)  - fp32 WMMA, HBM bound
//   phase 4: in-place log_softmax + argmax on d_out
// ---------------------------------------------------------------------------

#define EDIM 512
#define HDIM 1024
#define GDIM 3072   // 3*H
#define ZDIM 64
#define VDIM 50257
#define TSTEPS 64

typedef __attribute__((ext_vector_type(2))) float v2f;
typedef __attribute__((ext_vector_type(8))) float v8f;

// ---------------------------------------------------------------------------
// Phase 1: h0 = z @ W_dec + b_dec  (blocks 0..3, coalesced over j)
//          gi = W_ih @ eos + b_ih  (blocks 4..387, one wave per row)
// ---------------------------------------------------------------------------
__global__ void __launch_bounds__(256)
setup_kernel(const float* __restrict__ z, const float* __restrict__ Wdec,
             const float* __restrict__ bdec, const float* __restrict__ eos,
             const float* __restrict__ Wih, const float* __restrict__ bih,
             float* __restrict__ hcur, float* __restrict__ gi)
{
    if (blockIdx.x < 4) {
        const int j = blockIdx.x * 256 + threadIdx.x;      // 0..1023
        float acc = bdec[j];
        #pragma unroll 8
        for (int zi = 0; zi < ZDIM; ++zi)
            acc += z[zi] * Wdec[zi * HDIM + j];
        hcur[j] = acc;
    } else {
        const int lane = threadIdx.x & 31;
        const int wrow = (blockIdx.x - 4) * 8 + (threadIdx.x >> 5); // 0..3071
        const float4* w  = (const float4*)(Wih + (size_t)wrow * EDIM);
        const float4* e4 = (const float4*)eos;
        float acc = 0.f;
        #pragma unroll
        for (int i = 0; i < EDIM / 128; ++i) {             // 4 iters, coalesced
            float4 a = w[lane + 32 * i];
            float4 b = e4[lane + 32 * i];
            acc += a.x * b.x + a.y * b.y + a.z * b.z + a.w * b.w;
        }
        #pragma unroll
        for (int m = 16; m; m >>= 1) acc += __shfl_xor(acc, m, 32);
        if (lane == 0) gi[wrow] = acc + bih[wrow];
    }
}

// ---------------------------------------------------------------------------
// Phase 2a: gh = W_hh @ h + b_hh   (one wave per row, coalesced 512B/ld/wave)
// ---------------------------------------------------------------------------
__global__ void __launch_bounds__(256)
gru_gemv(const float* __restrict__ Whh, const float* __restrict__ bhh,
         const float* __restrict__ h, float* __restrict__ gh)
{
    const int lane = threadIdx.x & 31;
    const int wrow = blockIdx.x * 8 + (threadIdx.x >> 5);  // 0..3071
    const float4* w  = (const float4*)(Whh + (size_t)wrow * HDIM);
    const float4* hv = (const float4*)h;
    float acc = 0.f;
    #pragma unroll
    for (int i = 0; i < HDIM / 128; ++i) {                 // 8 iters
        float4 a = w[lane + 32 * i];
        float4 b = hv[lane + 32 * i];
        acc += a.x * b.x + a.y * b.y + a.z * b.z + a.w * b.w;
    }
    #pragma unroll
    for (int m = 16; m; m >>= 1) acc += __shfl_xor(acc, m, 32);
    if (lane == 0) gh[wrow] = acc + bhh[wrow];
}

// ---------------------------------------------------------------------------
// Phase 2b: gates + hidden update; store h_t into the H matrix row t
// ---------------------------------------------------------------------------
__global__ void __launch_bounds__(256)
gru_gate(const float* __restrict__ gi, const float* __restrict__ gh,
         float* __restrict__ hcur, float* __restrict__ Hmat, int t)
{
    const int j = blockIdx.x * 256 + threadIdx.x;          // 0..1023
    const float r = 1.f / (1.f + __expf(-(gi[j] + gh[j])));
    const float u = 1.f / (1.f + __expf(-(gi[HDIM + j] + gh[HDIM + j])));
    const float n = tanhf(gi[2 * HDIM + j] + r * gh[2 * HDIM + j]);
    const float hn = (1.f - u) * n + u * hcur[j];
    hcur[j] = hn;
    Hmat[(size_t)t * HDIM + j] = hn;
}

// ---------------------------------------------------------------------------
// Phase 3: logits = H(64x1024) @ W_vocab(1024xV) + b_vocab  via fp32 WMMA.
//
// Block = 128 threads = 4 waves.  Each WAVE owns one 16-wide n-tile
// (block covers 64 columns) and loops over all four 16-row m-tiles with
// 4 accumulators, so every B fragment (the HBM-bound stream) is loaded
// once and feeds 4 WMMAs.  A (64x1024 = 256 KB) is L2/L0 resident.
// The partial last tile is handled by CLAMPING load columns (no per-
// iteration EXEC masking) and guarding only the store.
//
// fp32 operand layouts (ISA 7.12.2):
//   A 16x4 : lanes 0-15 -> M=lane, VGPR0={K,K+1} ; lanes 16-31 -> {K+2,K+3}
//   D 16x16: VGPR r -> M=r (lanes 0-15) / M=8+r (lanes 16-31), N=lane&15
// ---------------------------------------------------------------------------
__global__ void __launch_bounds__(128)
vocab_gemm_wmma(const float* __restrict__ Hmat, const float* __restrict__ Wv,
                const float* __restrict__ bv, float* __restrict__ logits)
{
    const int lane   = threadIdx.x & 31;
    const int wave   = threadIdx.x >> 5;                 // 0..3 -> n tile
    const int n_base = blockIdx.x * 64 + wave * 16;
    const int ncol   = lane & 15;
    const int khi    = lane >> 4;                        // 0:{K,K+1} 1:{K+2,K+3}

    const int n   = n_base + ncol;
    const int ncl = (n < VDIM) ? n : (VDIM - 1);         // clamp for loads
    const float* bcol = Wv + (size_t)ncl;

    // A row pointers for the 4 m-tiles (row = 16*mt + ncol)
    const float* a0 = Hmat + (size_t)ncol * HDIM + 2 * khi;
    const float* a1 = a0 + (size_t)16 * HDIM;
    const float* a2 = a0 + (size_t)32 * HDIM;
    const float* a3 = a0 + (size_t)48 * HDIM;

    v8f c0 = {0.f,0.f,0.f,0.f,0.f,0.f,0.f,0.f};
    v8f c1 = c0, c2 = c0, c3 = c0;

    #pragma unroll 4
    for (int k = 0; k < HDIM; k += 4) {
        v2f b;
        b.x = bcol[(size_t)(k + khi) * VDIM];
        b.y = bcol[(size_t)(k + 2 + khi) * VDIM];
        v2f x0 = *(const v2f*)(a0 + k);
        v2f x1 = *(const v2f*)(a1 + k);
        v2f x2 = *(const v2f*)(a2 + k);
        v2f x3 = *(const v2f*)(a3 + k);
        c0 = __builtin_amdgcn_wmma_f32_16x16x4_f32(false, x0, false, b,
                                                   (short)0, c0, false, false);
        c1 = __builtin_amdgcn_wmma_f32_16x16x4_f32(false, x1, false, b,
                                                   (short)0, c1, false, false);
        c2 = __builtin_amdgcn_wmma_f32_16x16x4_f32(false, x2, false, b,
                                                   (short)0, c2, false, false);
        c3 = __builtin_amdgcn_wmma_f32_16x16x4_f32(false, x3, false, b,
                                                   (short)0, c3, false, false);
    }

    if (n < VDIM) {
        const float bias = bv[n];
        const int   msub = khi << 3;                     // +8 for lanes 16-31
        float* o = logits + (size_t)msub * VDIM + n;
        #pragma unroll
        for (int r = 0; r < 8; ++r) {
            o[(size_t)(r)      * VDIM] = c0[r] + bias;   // m-tile 0
            o[(size_t)(16 + r) * VDIM] = c1[r] + bias;   // m-tile 1
            o[(size_t)(32 + r) * VDIM] = c2[r] + bias;   // m-tile 2
            o[(size_t)(48 + r) * VDIM] = c3[r] + bias;   // m-tile 3
        }
    }
}

// ---------------------------------------------------------------------------
// Phase 4: in-place log_softmax + argmax per row of d_out.
// One block per time step; online (max,sum) + (val,idx) reduction.
// ---------------------------------------------------------------------------
__global__ void __launch_bounds__(256)
softmax_argmax(float* __restrict__ out)
{
    const int t   = blockIdx.x;
    const int tid = threadIdx.x;
    float* row = out + (size_t)t * VDIM;

    float m = -INFINITY, s = 0.f, bval = -INFINITY;
    int   bidx = 0;
    for (int i = tid; i < VDIM; i += 256) {
        const float x = row[i];
        if (x > bval) { bval = x; bidx = i; }
        if (x > m) { s = s * __expf(m - x) + 1.f; m = x; }
        else       { s += __expf(x - m); }
    }

    __shared__ float sm[256], ss[256], sv[256];
    __shared__ int   si[256];
    sm[tid] = m; ss[tid] = s; sv[tid] = bval; si[tid] = bidx;
    __syncthreads();
    for (int off = 128; off; off >>= 1) {
        if (tid < off) {
            const float m2 = sm[tid + off], s2 = ss[tid + off];
            if (m2 > sm[tid]) { ss[tid] = ss[tid] * __expf(sm[tid] - m2) + s2; sm[tid] = m2; }
            else              { ss[tid] += s2 * __expf(m2 - sm[tid]); }
            const float v2 = sv[tid + off]; const int i2 = si[tid + off];
            if (v2 > sv[tid] || (v2 == sv[tid] && i2 < si[tid])) { sv[tid] = v2; si[tid] = i2; }
        }
        __syncthreads();
    }
    const float logZ = sm[0] + __logf(ss[0]);
    if (tid == 0)
        ((int*)out)[(size_t)TSTEPS * VDIM + t] = si[0];   // seq (int32 bits)
    for (int i = tid; i < VDIM; i += 256)
        row[i] = row[i] - logZ;                            // log_p
}

// ---------------------------------------------------------------------------
extern "C" void kernel_launch(void* const* d_in, const int* in_sizes, int n_in,
                              void* d_out, int out_size, void* d_ws, size_t ws_size,
                              hipStream_t stream)
{
    const float* z      = (const float*)d_in[0];
    const float* eos    = (const float*)d_in[1];
    const float* W_ih   = (const float*)d_in[2];
    const float* W_hh   = (const float*)d_in[3];
    const float* b_ih   = (const float*)d_in[4];
    const float* b_hh   = (const float*)d_in[5];
    const float* W_dec  = (const float*)d_in[6];
    const float* b_dec  = (const float*)d_in[7];
    const float* W_voc  = (const float*)d_in[8];
    const float* b_voc  = (const float*)d_in[9];
    float* out = (float*)d_out;

    // workspace layout (floats): hcur | gi | gh | Hmat  (~284 KB total)
    float* ws   = (float*)d_ws;
    float* hcur = ws;
    float* gi   = ws + 1024;
    float* gh   = ws + 1024 + 3072;
    float* Hmat = ws + 1024 + 3072 + 3072;

    // phase 1
    setup_kernel<<<4 + GDIM / 8, 256, 0, stream>>>(z, W_dec, b_dec, eos,
                                                   W_ih, b_ih, hcur, gi);
    // phase 2: serial recurrence (graph replay amortizes the 128 launches)
    for (int t = 0; t < TSTEPS; ++t) {
        gru_gemv<<<GDIM / 8, 256, 0, stream>>>(W_hh, b_hh, hcur, gh);
        gru_gate<<<HDIM / 256, 256, 0, stream>>>(gi, gh, hcur, Hmat, t);
    }
    // phase 3: the big WMMA GEMM (W_vocab streamed exactly once)
    const int nblocks = (VDIM + 63) / 64;   // 786, each covers 64 columns
    vocab_gemm_wmma<<<nblocks, 128, 0, stream>>>(Hmat, W_voc, b_voc, out);
    // phase 4
    softmax_argmax<<<TSTEPS, 256, 0, stream>>>(out);
}